// QIDDM_CL_new_21285857919126
// MI455X (gfx1250) — compile-verified
//
#include <hip/hip_runtime.h>
#include <math.h>

typedef float v2f __attribute__((ext_vector_type(2)));
typedef float v8f __attribute__((ext_vector_type(8)));

#define QW   12
#define DIM  4096
#define IMGS 32

// ---------------------------------------------------------------------------
// Kernel 1: build the 36 grouped gate matrices (16x16 complex each).
// kmat layout: [n(2)][i(3)][l(2)][g(3)] -> 512 floats = 256 Re then 256 Im.
// K[m][k] = prod_{q=0..3} U_{wire=g*4+q}[ bit(m,3-q), bit(k,3-q) ]
// U = Rot(phi,theta,omega) = RZ(omega) RY(theta) RZ(phi)
// ---------------------------------------------------------------------------
__global__ void k_gatemats(const float* __restrict__ w, float* __restrict__ kmat) {
    int id = blockIdx.x;                 // n*18 + i*6 + l*3 + g
    int g = id % 3;
    int l = (id / 3) % 2;
    int i = (id / 6) % 3;
    int n = id / 18;
    int t = threadIdx.x;                 // 256 threads: one matrix entry each
    int m = t >> 4, k = t & 15;

    float pr = 1.f, pi = 0.f;
    for (int q = 0; q < 4; ++q) {
        int wire = g * 4 + q;
        const float* wp = w + (((n * 3 + i) * 2 + l) * QW + wire) * 3;
        float phi = wp[0], th = wp[1], om = wp[2];
        float s, c;  sincosf(0.5f * th, &s, &c);
        float sa, ca; sincosf(0.5f * (phi + om), &sa, &ca);
        float sb, cb; sincosf(0.5f * (phi - om), &sb, &cb);
        int mb = (m >> (3 - q)) & 1;
        int kb = (k >> (3 - q)) & 1;
        float ur, ui;
        if (mb == 0 && kb == 0)      { ur =  ca * c; ui = -sa * c; }  // e^{-ia} c
        else if (mb == 0)            { ur = -cb * s; ui = -sb * s; }  // -e^{+ib} s
        else if (kb == 0)            { ur =  cb * s; ui = -sb * s; }  // e^{-ib} s
        else                         { ur =  ca * c; ui =  sa * c; }  // e^{+ia} c
        float nr = pr * ur - pi * ui;
        float ni = pr * ui + pi * pr * 0.f + (pi * ur);  // keep exact form below
        ni = pr * ui + pi * ur;
        pr = nr; pi = ni;
    }
    float* base = kmat + id * 512;
    base[t]       = pr;
    base[256 + t] = pi;
}

// ---------------------------------------------------------------------------
// Kernel 2: conv(3x3, stride2, pad1) + bias + global average pool -> angles(b,12)
// ---------------------------------------------------------------------------
__global__ void k_convpool(const float* __restrict__ x, const float* __restrict__ cw,
                           const float* __restrict__ cb, float* __restrict__ ang) {
    __shared__ float part[256 * QW];
    int b = blockIdx.x, t = threadIdx.x;     // 256 threads = 16x16 output positions
    int oh = t >> 4, ow = t & 15;
    const float* xi = x + (size_t)b * IMGS * IMGS;

    float patch[9];
    for (int kh = 0; kh < 3; ++kh)
        for (int kw = 0; kw < 3; ++kw) {
            int ih = oh * 2 - 1 + kh;
            int iw = ow * 2 - 1 + kw;
            patch[kh * 3 + kw] = (ih >= 0 && ih < IMGS && iw >= 0 && iw < IMGS)
                                     ? xi[ih * IMGS + iw] : 0.f;
        }
    for (int c = 0; c < QW; ++c) {
        float s = 0.f;
        for (int kk = 0; kk < 9; ++kk) s = fmaf(patch[kk], cw[c * 9 + kk], s);
        part[t * QW + c] = s;
    }
    __syncthreads();
    if (t < QW) {
        float s = 0.f;
        for (int kq = 0; kq < 256; ++kq) s += part[kq * QW + t];   // deterministic
        ang[b * QW + t] = s * (1.f / 256.f) + cb[t];
    }
}

// ---------------------------------------------------------------------------
// Kernel 3: BOTH circuit blocks fused. One workgroup (128 thr = 4 waves) per
// sample. State (4096 cplx) + encoding phases live in LDS (64KB); inter-block
// expvals are reduced in LDS and re-broadcast, never touching HBM.
// Sublayer unitary = 3 tensor-mode contractions with 16x16 complex matrices
// via V_WMMA_F32_16X16X4_F32 (full fp32). All inner loops unrolled so the
// address math is compile-time affine and control flow is uniform (EXEC=all-1
// around every WMMA).
// ---------------------------------------------------------------------------
__global__ void __launch_bounds__(128)
k_circuit2(const float* __restrict__ angIn, float* __restrict__ angOut,
           const float* __restrict__ kmatAll) {
    __shared__ float ls[16384];                 // 64 KB
    float* stRe  = ls;
    float* stIm  = ls + 4096;
    float* encRe = ls + 8192;                   // also reused as reduction scratch
    float* encIm = ls + 12288;                  // ls+12288..+12299 = broadcast slot

    int b = blockIdx.x, t = threadIdx.x;
    int lane = t & 31, wv = t >> 5;
    int mhalf = (lane >> 4) * 8;   // C/D row offset for this lane half
    int khalf = (lane >> 4) * 2;   // A/B K offset within each k-tile
    int nidx  = lane & 15;         // column (and A row) index

    float xj[QW];
    for (int j = 0; j < QW; ++j) xj[j] = angIn[b * QW + j];

    for (int n = 0; n < 2; ++n) {
        const float* kmat = kmatAll + n * 18 * 512;

        // encoding phases exp(i * sum_j x_j*(bit_j - 0.5)) and |0> init
        for (int i = t; i < DIM; i += 128) {
            float ph = 0.f;
            for (int j = 0; j < QW; ++j) {
                float bit = (float)((i >> (11 - j)) & 1);
                ph = fmaf(xj[j], bit - 0.5f, ph);
            }
            float si, co; sincosf(ph, &si, &co);
            encRe[i] = co; encIm[i] = si;
            stRe[i] = (i == 0) ? 1.f : 0.f;
            stIm[i] = 0.f;
        }
        __syncthreads();

        for (int layer = 0; layer < 3; ++layer) {
            // st *= enc
            for (int i = t; i < DIM; i += 128) {
                float ar = stRe[i], ai = stIm[i];
                float er = encRe[i], ei = encIm[i];
                stRe[i] = ar * er - ai * ei;
                stIm[i] = ar * ei + ai * er;
            }
            __syncthreads();

#pragma unroll
            for (int l = 0; l < 2; ++l) {
#pragma unroll
                for (int g = 0; g < 3; ++g) {
                    // per-mode constant strides: element (k, n) of tile tl sits
                    // at base(tl,n) + k*SK in the 4096-element state.
                    const int SK = (g == 0) ? 256 : (g == 1) ? 16 : 1;

                    const float* Kre = kmat + ((layer * 2 + l) * 3 + g) * 512;
                    const float* Kim = Kre + 256;

                    // A operand (gate matrix), same for all tiles of this wave
                    v2f A_r[4], A_i[4];
#pragma unroll
                    for (int kt = 0; kt < 4; ++kt) {
                        int kk = kt * 4 + khalf;
                        v2f ar, ai;
                        ar[0] = Kre[nidx * 16 + kk];  ar[1] = Kre[nidx * 16 + kk + 1];
                        ai[0] = Kim[nidx * 16 + kk];  ai[1] = Kim[nidx * 16 + kk + 1];
                        A_r[kt] = ar; A_i[kt] = ai;
                    }

#pragma unroll
                    for (int tt = 0; tt < 4; ++tt) {
                        int tl = wv * 4 + tt;
                        int base = (g == 0) ? (tl * 16 + nidx)
                                 : (g == 1) ? (tl * 256 + nidx)
                                            : (tl * 256 + nidx * 16);
                        v2f B_r[4], B_i[4];
#pragma unroll
                        for (int kt = 0; kt < 4; ++kt) {
                            int i0 = base + (kt * 4 + khalf) * SK;
                            v2f br, bi;
                            br[0] = stRe[i0]; br[1] = stRe[i0 + SK];
                            bi[0] = stIm[i0]; bi[1] = stIm[i0 + SK];
                            B_r[kt] = br; B_i[kt] = bi;
                        }
                        v8f p1 = {}; v8f p2 = {}; v8f p3 = {}; v8f p4 = {};
#pragma unroll
                        for (int kt = 0; kt < 4; ++kt) {
                            p1 = __builtin_amdgcn_wmma_f32_16x16x4_f32(false, A_r[kt], false, B_r[kt], (short)0, p1, false, false);
                            p2 = __builtin_amdgcn_wmma_f32_16x16x4_f32(false, A_i[kt], false, B_i[kt], (short)0, p2, false, false);
                            p3 = __builtin_amdgcn_wmma_f32_16x16x4_f32(false, A_r[kt], false, B_i[kt], (short)0, p3, false, false);
                            p4 = __builtin_amdgcn_wmma_f32_16x16x4_f32(false, A_i[kt], false, B_r[kt], (short)0, p4, false, false);
                        }
#pragma unroll
                        for (int r = 0; r < 8; ++r) {
                            int idx = base + (r + mhalf) * SK;
                            stRe[idx] = p1[r] - p2[r];
                            stIm[idx] = p3[r] + p4[r];
                        }
                    }
                    __syncthreads();
                }
                // CZ ring sign for range r = l+1 (l unrolled -> constant shifts)
                const int rr = l + 1;
                for (int i = t; i < DIM; i += 128) {
                    int rot = ((i << rr) | (i >> (12 - rr))) & 0xFFF;
                    if (__popc(i & rot) & 1) { stRe[i] = -stRe[i]; stIm[i] = -stIm[i]; }
                }
                __syncthreads();
            }
        }

        // expectation values <Z_j> = sum_s |st_s|^2 * (1 - 2*bit_j(s))
        float acc[QW];
#pragma unroll
        for (int j = 0; j < QW; ++j) acc[j] = 0.f;
        for (int i = t; i < DIM; i += 128) {
            float p = stRe[i] * stRe[i] + stIm[i] * stIm[i];
#pragma unroll
            for (int j = 0; j < QW; ++j)
                acc[j] += ((i >> (11 - j)) & 1) ? -p : p;
        }
        __syncthreads();
        float* part = encRe;                    // enc region no longer needed
#pragma unroll
        for (int j = 0; j < QW; ++j) part[t * QW + j] = acc[j];
        __syncthreads();
        if (t < QW) {
            float s = 0.f;
            for (int kq = 0; kq < 128; ++kq) s += part[kq * QW + t]; // deterministic
            encIm[t] = s;                       // broadcast slot
        }
        __syncthreads();
#pragma unroll
        for (int j = 0; j < QW; ++j) xj[j] = encIm[j];  // all threads reload
        __syncthreads();
    }

    if (t < QW) angOut[b * QW + t] = xj[t];
}

// ---------------------------------------------------------------------------
// Kernel 4: out[b,k] = lin_b[k] + sum_j angles[b,j] * lin_w[k,j]
// ---------------------------------------------------------------------------
__global__ void k_linear(const float* __restrict__ ang, const float* __restrict__ lw,
                         const float* __restrict__ lb, float* __restrict__ out, int total) {
    int idx = blockIdx.x * 256 + threadIdx.x;
    if (idx >= total) return;
    int b = idx >> 10, k = idx & 1023;
    float s = lb[k];
    const float* a = ang + b * QW;
    const float* wr = lw + k * QW;
    for (int j = 0; j < QW; ++j) s = fmaf(a[j], wr[j], s);
    out[idx] = s;
}

// ---------------------------------------------------------------------------
extern "C" void kernel_launch(void* const* d_in, const int* in_sizes, int n_in,
                              void* d_out, int out_size, void* d_ws, size_t ws_size,
                              hipStream_t stream) {
    const float* x  = (const float*)d_in[0];   // (B,1,32,32)
    const float* cw = (const float*)d_in[1];   // (12,1,3,3)
    const float* cb = (const float*)d_in[2];   // (12,)
    const float* w1 = (const float*)d_in[3];   // (2,3,2,12,3)
    const float* lw = (const float*)d_in[4];   // (1024,12)
    const float* lb = (const float*)d_in[5];   // (1024,)
    float* out = (float*)d_out;

    int B = in_sizes[0] / (IMGS * IMGS);

    float* kmat = (float*)d_ws;                // 36 * 512 floats
    float* angA = kmat + 36 * 512;             // B*12  (conv output)
    float* angC = angA + B * QW;               // B*12  (after both blocks)

    k_gatemats<<<36, 256, 0, stream>>>(w1, kmat);
    k_convpool<<<B, 256, 0, stream>>>(x, cw, cb, angA);
    k_circuit2<<<B, 128, 0, stream>>>(angA, angC, kmat);
    int total = B * IMGS * IMGS;
    k_linear<<<(total + 255) / 256, 256, 0, stream>>>(angC, lw, lb, out, total);
}